// Dubois_65970697666733
// MI455X (gfx1250) — compile-verified
//
#include <hip/hip_runtime.h>
#include <hip/hip_bf16.h>

// Output layout per row: [16 singles][120 pairs][560 triples] = 696 columns.
#define NCOLS 696
#define NIT   22            // ceil(696 / 32) column-iterations per wave
#define NPAD  24            // per-lane entries padded to 96 B (6 x b128)

// Packed combo entry: three pre-scaled ds_bpermute byte indices.
// Lane 16 holds the constant 1.0f, and T(h, 1) == h exactly, so:
//   singles -> (i, 16, 16), pairs -> (i, j, 16), triples -> (i, j, k).
// Every column runs the identical branchless 2-fold body.
struct alignas(16) Combos { unsigned int v[32][NPAD]; };

constexpr unsigned int packc(int i, int j, int k) {
    return (unsigned)(i * 4) | ((unsigned)(j * 4) << 8) | ((unsigned)(k * 4) << 16);
}

constexpr Combos makeCombos() {
    unsigned int lin[NIT * 32] = {};
    int c = 0;
    for (int i = 0; i < 16; ++i)
        lin[c++] = packc(i, 16, 16);
    for (int i = 0; i < 16; ++i)
        for (int j = i + 1; j < 16; ++j)
            lin[c++] = packc(i, j, 16);
    for (int i = 0; i < 16; ++i)
        for (int j = i + 1; j < 16; ++j)
            for (int k = j + 1; k < 16; ++k)
                lin[c++] = packc(i, j, k);
    for (; c < NIT * 32; ++c)
        lin[c] = packc(16, 16, 16);          // tail pads: compute 1.0, never stored

    // Transpose: lane-major so each lane's descriptors are contiguous.
    Combos t{};
    for (int lane = 0; lane < 32; ++lane) {
        for (int it = 0; it < NIT; ++it)
            t.v[lane][it] = lin[lane + 32 * it];
        for (int it = NIT; it < NPAD; ++it)
            t.v[lane][it] = packc(16, 16, 16);
    }
    return t;
}

__constant__ Combos g_combos = makeCombos();

// Dubois t-norm: a*b / max(max(a,b), lam). Denominator >= lam = 0.3,
// fast v_rcp_f32 (~1 ulp) is plenty for fp32 tolerances; rcp(1.0)==1.0
// exactly, which the sentinel-lane identity relies on.
__device__ __forceinline__ float dubois_t(float a, float b, float lam) {
    float m = fmaxf(fmaxf(a, b), lam);       // fuses to v_max3_num_f32
    return a * b * __builtin_amdgcn_rcpf(m);
}

__global__ __launch_bounds__(256) void dubois_kernel(
    const float* __restrict__ x, const float* __restrict__ lambda_p,
    float* __restrict__ out, int nrows) {
    const int lane   = threadIdx.x & 31;
    const int wave   = (int)((blockIdx.x * blockDim.x + threadIdx.x) >> 5);
    const int nwaves = (int)((gridDim.x * blockDim.x) >> 5);
    const float lam  = lambda_p[0];                    // uniform scalar load
    const bool  low16 = (lane < 16);
    const bool  tailOn = (lane < (NCOLS & 31));        // lanes 0..23 in iter 21

    // One-time per-lane descriptor fetch: 6 contiguous b128 loads.
    unsigned int combo[NPAD];
    {
        const uint4* lrow = (const uint4*)g_combos.v[lane];
#pragma unroll
        for (int q = 0; q < NPAD / 4; ++q) {
            uint4 d = lrow[q];
            combo[4 * q + 0] = d.x;
            combo[4 * q + 1] = d.y;
            combo[4 * q + 2] = d.z;
            combo[4 * q + 3] = d.w;
        }
    }

    for (int row = wave; row < nrows; row += nwaves) {
        // Lanes 0..15: the 16 row values (coalesced 64B read).
        // Lanes 16..31: identity element 1.0f for the sentinel trick.
        const float xv  = x[(size_t)row * 16 + (lane & 15)];
        const float val = low16 ? xv : 1.0f;
        const int   vi  = __float_as_int(val);
        float* outRow = out + (size_t)row * NCOLS;

#pragma unroll
        for (int it = 0; it < NIT; ++it) {
            const unsigned v  = combo[it];
            const int ia = v & 0xFF;             // pre-scaled byte indices
            const int ib = (v >> 8) & 0xFF;
            const int ic = (v >> 16) & 0xFF;
            float a  = __int_as_float(__builtin_amdgcn_ds_bpermute(ia, vi));
            float b  = __int_as_float(__builtin_amdgcn_ds_bpermute(ib, vi));
            float c2 = __int_as_float(__builtin_amdgcn_ds_bpermute(ic, vi));
            float h = dubois_t(dubois_t(a, b, lam), c2, lam);  // branchless
            // Store predicate resolves at compile time except the last iter.
            if (32 * (it + 1) <= NCOLS) {
                __builtin_nontemporal_store(h, outRow + lane + 32 * it);
            } else if (tailOn) {
                __builtin_nontemporal_store(h, outRow + lane + 32 * it);
            }
        }
    }
}

extern "C" void kernel_launch(void* const* d_in, const int* in_sizes, int n_in,
                              void* d_out, int out_size, void* d_ws, size_t ws_size,
                              hipStream_t stream) {
    const float* x   = (const float*)d_in[0];
    const float* lam = (const float*)d_in[1];
    float* out       = (float*)d_out;
    const int nrows  = in_sizes[0] / 16;

    const int threads = 256;                 // 8 waves per block (wave32)
    const int wavesPerBlock = threads / 32;
    int blocks = (nrows + wavesPerBlock - 1) / wavesPerBlock;
    if (blocks > 2048) blocks = 2048;        // 8 rows/wave amortizes setup

    dubois_kernel<<<blocks, threads, 0, stream>>>(x, lam, out, nrows);
}